// Model_87230785782112
// MI455X (gfx1250) — compile-verified
//
#include <hip/hip_runtime.h>

#define N_USER 200000
#define N_DA   100000
#define N_DB   100000
#define NE     2000000
#define D      64
#define WSTRIDE 68   // padded LDS stride: (4m+2kh) pattern -> conflict-free banks

typedef __attribute__((ext_vector_type(2))) float v2f;
typedef __attribute__((ext_vector_type(4))) float v4f;
typedef __attribute__((ext_vector_type(8))) float v8f;

// ---------------------------------------------------------------------------
// Zero-fill (grid-stride)
// ---------------------------------------------------------------------------
__global__ void fill_zero_kernel(float* __restrict__ p, int n) {
    int i = blockIdx.x * blockDim.x + threadIdx.x;
    int stride = gridDim.x * blockDim.x;
    for (; i < n; i += stride) p[i] = 0.0f;
}

// ---------------------------------------------------------------------------
// In-degree: one thread per edge, hardware f32 atomic into deg[dst[e]]
// ---------------------------------------------------------------------------
__global__ void degree_kernel(const int* __restrict__ dst,
                              float* __restrict__ deg, int ne) {
    int e = blockIdx.x * blockDim.x + threadIdx.x;
    if (e < ne) unsafeAtomicAdd(&deg[dst[e]], 1.0f);
}

// ---------------------------------------------------------------------------
// Edge scatter: 16 threads per edge, each moves a float4 (coalesced 256B/edge)
// acc[dst] += x_src[src]   via global_atomic_add_f32 (L2-resident working set)
// ---------------------------------------------------------------------------
__global__ void scatter_kernel(const float* __restrict__ xsrc,
                               const int* __restrict__ src,
                               const int* __restrict__ dst,
                               float* __restrict__ acc, int ne) {
    int tid = blockIdx.x * blockDim.x + threadIdx.x;
    int e = tid >> 4;
    int q = tid & 15;
    if (e >= ne) return;
    int s = src[e];
    int d = dst[e];
    v4f v = *(const v4f*)(xsrc + (size_t)s * D + q * 4);
    float* o = acc + (size_t)d * D + q * 4;
    unsafeAtomicAdd(o + 0, v.x);
    unsafeAtomicAdd(o + 1, v.y);
    unsafeAtomicAdd(o + 2, v.z);
    unsafeAtomicAdd(o + 3, v.w);
}

// ---------------------------------------------------------------------------
// Normalize + [16xD] @ [DxD] + bias via V_WMMA_F32_16X16X4_F32.
// W is staged once per block into LDS, transposed with padded stride so each
// B fragment is a single conflict-free ds_load_b64.
//
// A frag (16x4 f32, ISA 7.12.2):  lanes 0-15: M=lane,  K=k0+{0,1}
//                                 lanes16-31: M=lane-16,K=k0+{2,3}
// B frag (4x16): VGPR v holds K=k0+2*kh+v, N=lane&15 (+16*ntile)
// C/D (16x16 f32): VGPR r: lanes 0-15 -> (M=r, N=lane); lanes 16-31 -> (M=8+r)
// ---------------------------------------------------------------------------
__global__ void __launch_bounds__(256)
gemm_wmma_kernel(const float* __restrict__ acc,
                 const float* __restrict__ deg,
                 const float* __restrict__ W,
                 const float* __restrict__ bias,
                 float* __restrict__ out,
                 int nrows, int accumulate) {
    __shared__ float sWt[D * WSTRIDE];   // sWt[n][k] = W[k][n], stride WSTRIDE
    __shared__ float sBias[D];

    const int tid = threadIdx.x;

    // Cooperative stage: coalesced global read, transposed LDS write.
    for (int i = tid; i < D * D; i += 256) {
        const int k = i >> 6;
        const int n = i & 63;
        sWt[n * WSTRIDE + k] = W[i];
    }
    if (tid < D) sBias[tid] = bias[tid];
    __syncthreads();

    const int lane = tid & 31;
    const int wave = tid >> 5;
    const int ntiles = nrows >> 4;
    const int tile = blockIdx.x * 8 + wave;
    if (tile >= ntiles) return;            // wave-uniform: EXEC stays all-1s
    const int rowBase = tile << 4;
    const int m  = lane & 15;              // row-in-tile / N-in-tile
    const int kh = lane >> 4;              // which K-half this lane holds
    const int row = rowBase + m;

    const float invd = 1.0f / fmaxf(deg[row], 1.0f);
    const float* arow = acc + (size_t)row * D;

    const float* w0p = &sWt[(m)      * WSTRIDE];
    const float* w1p = &sWt[(16 + m) * WSTRIDE];
    const float* w2p = &sWt[(32 + m) * WSTRIDE];
    const float* w3p = &sWt[(48 + m) * WSTRIDE];

    v8f c0 = {}, c1 = {}, c2 = {}, c3 = {};

#pragma unroll
    for (int kk = 0; kk < 16; ++kk) {
        const int k0 = (kk << 2) + (kh << 1);
        v2f a;
        a.x = arow[k0]     * invd;
        a.y = arow[k0 + 1] * invd;

        v2f b0 = *(const v2f*)(w0p + k0);   // ds_load_b64, conflict-free banks
        v2f b1 = *(const v2f*)(w1p + k0);
        v2f b2 = *(const v2f*)(w2p + k0);
        v2f b3 = *(const v2f*)(w3p + k0);

        c0 = __builtin_amdgcn_wmma_f32_16x16x4_f32(false, a, false, b0, (short)0, c0, false, false);
        c1 = __builtin_amdgcn_wmma_f32_16x16x4_f32(false, a, false, b1, (short)0, c1, false, false);
        c2 = __builtin_amdgcn_wmma_f32_16x16x4_f32(false, a, false, b2, (short)0, c2, false, false);
        c3 = __builtin_amdgcn_wmma_f32_16x16x4_f32(false, a, false, b3, (short)0, c3, false, false);
    }

#pragma unroll
    for (int r = 0; r < 8; ++r) {
        const int orow = rowBase + (kh << 3) + r;
        float* o = out + (size_t)orow * D + m;
        float t0 = c0[r] + sBias[m];
        float t1 = c1[r] + sBias[16 + m];
        float t2 = c2[r] + sBias[32 + m];
        float t3 = c3[r] + sBias[48 + m];
        if (accumulate) {
            o[0]  += t0; o[16] += t1; o[32] += t2; o[48] += t3;
        } else {
            o[0]  = t0;  o[16] = t1;  o[32] = t2;  o[48] = t3;
        }
    }
}

// ---------------------------------------------------------------------------
extern "C" void kernel_launch(void* const* d_in, const int* in_sizes, int n_in,
                              void* d_out, int out_size, void* d_ws, size_t ws_size,
                              hipStream_t stream) {
    const float* x_user = (const float*)d_in[0];
    const float* x_da   = (const float*)d_in[1];
    const float* x_db   = (const float*)d_in[2];
    const float* W1     = (const float*)d_in[3];
    const float* b1     = (const float*)d_in[4];
    const float* W2     = (const float*)d_in[5];
    const float* b2     = (const float*)d_in[6];
    const int* src_u2a  = (const int*)d_in[7];
    const int* dst_u2a  = (const int*)d_in[8];
    const int* src_a2u  = (const int*)d_in[9];
    const int* dst_a2u  = (const int*)d_in[10];
    const int* src_u2b  = (const int*)d_in[11];
    const int* dst_u2b  = (const int*)d_in[12];
    const int* src_b2u  = (const int*)d_in[13];
    const int* dst_b2u  = (const int*)d_in[14];

    // workspace layout (floats)
    float* ws       = (float*)d_ws;
    float* deg_u2a  = ws;                                   // [N_DA]
    float* deg_a2u  = deg_u2a + N_DA;                       // [N_USER]
    float* deg_u2b  = deg_a2u + N_USER;                     // [N_DB]
    float* deg_b2u  = deg_u2b + N_DB;                       // [N_USER]
    float* acc      = deg_b2u + N_USER;                     // [N_USER*D] (max dst)
    float* xu1      = acc + (size_t)N_USER * D;             // [N_USER*D]
    float* xa1      = xu1 + (size_t)N_USER * D;             // [N_DA*D]
    float* xb1      = xa1 + (size_t)N_DA * D;               // [N_DB*D]

    float* out_u = (float*)d_out;                           // xu2
    float* out_a = out_u + (size_t)N_USER * D;              // xa2
    float* out_b = out_a + (size_t)N_DA * D;                // xb2

    const int ZB = 256;

    // ---- degrees (edge structure only; recomputed each call, deterministic)
    const int degTotal = N_DA + N_USER + N_DB + N_USER;
    fill_zero_kernel<<<2048, ZB, 0, stream>>>(deg_u2a, degTotal);
    const int degGrid = (NE + ZB - 1) / ZB;
    degree_kernel<<<degGrid, ZB, 0, stream>>>(dst_u2a, deg_u2a, NE);
    degree_kernel<<<degGrid, ZB, 0, stream>>>(dst_a2u, deg_a2u, NE);
    degree_kernel<<<degGrid, ZB, 0, stream>>>(dst_u2b, deg_u2b, NE);
    degree_kernel<<<degGrid, ZB, 0, stream>>>(dst_b2u, deg_b2u, NE);

    auto relation = [&](const float* xsrc, const int* s, const int* dd, int ndst,
                        const float* W, const float* bb, const float* deg,
                        float* out, int accum) {
        fill_zero_kernel<<<4096, ZB, 0, stream>>>(acc, ndst * D);
        const long long nScat = (long long)NE * 16;
        scatter_kernel<<<(int)((nScat + ZB - 1) / ZB), ZB, 0, stream>>>(xsrc, s, dd, acc, NE);
        const int ntiles = ndst / 16;            // exact: all counts % 16 == 0
        gemm_wmma_kernel<<<(ntiles + 7) / 8, ZB, 0, stream>>>(acc, deg, W, bb, out, ndst, accum);
    };

    // ---- layer 1 -> intermediates in ws
    relation(x_user, src_u2a, dst_u2a, N_DA,   W1 + 0 * D * D, b1 + 0 * D, deg_u2a, xa1, 0);
    relation(x_da,   src_a2u, dst_a2u, N_USER, W1 + 1 * D * D, b1 + 1 * D, deg_a2u, xu1, 0);
    relation(x_user, src_u2b, dst_u2b, N_DB,   W1 + 2 * D * D, b1 + 2 * D, deg_u2b, xb1, 0);
    relation(x_db,   src_b2u, dst_b2u, N_USER, W1 + 3 * D * D, b1 + 3 * D, deg_b2u, xu1, 1);

    // ---- layer 2 -> d_out
    relation(xu1, src_u2a, dst_u2a, N_DA,   W2 + 0 * D * D, b2 + 0 * D, deg_u2a, out_a, 0);
    relation(xa1, src_a2u, dst_a2u, N_USER, W2 + 1 * D * D, b2 + 1 * D, deg_a2u, out_u, 0);
    relation(xu1, src_u2b, dst_u2b, N_DB,   W2 + 2 * D * D, b2 + 2 * D, deg_u2b, out_b, 0);
    relation(xb1, src_b2u, dst_b2u, N_USER, W2 + 3 * D * D, b2 + 3 * D, deg_b2u, out_u, 1);
}